// ZPNN_CorrelationLoss_84155589198583
// MI455X (gfx1250) — compile-verified
//
#include <hip/hip_runtime.h>
#include <stdint.h>
#include <math.h>

#define IMG_H 2048
#define IMG_W 2048
#define EPSF  1e-20f

typedef __attribute__((ext_vector_type(2))) float        v2f;
typedef __attribute__((ext_vector_type(8))) float        v8f;
typedef __attribute__((ext_vector_type(4))) unsigned int u32x4;
typedef __attribute__((ext_vector_type(8))) int          i32x8;
typedef __attribute__((ext_vector_type(4))) int          i32x4;

// ---------------------------------------------------------------------------
// f32 WMMA, 16x16x4: D = A(16x4) * B(4x16) + C.  Full f32 precision, which
// matches the reference conv path exactly (conv is f32 in the reference).
// ---------------------------------------------------------------------------
__device__ __forceinline__ v8f wmma_f32_k4(v2f a, v2f b, v8f c) {
  return __builtin_amdgcn_wmma_f32_16x16x4_f32(
      false, a, false, b, (short)0, c, false, false);
}

__device__ __forceinline__ uint32_t lds_off_of(const void* p) {
  // LDS->flat addrspace cast puts the workgroup-relative LDS offset in the
  // low 32 bits (ISA: LDS aperture address = {shared_base, offset}).
  return (uint32_t)(uint64_t)(uintptr_t)p;
}

// ---------------------------------------------------------------------------
// TDM: async DMA of a 2D f32 tile (tile_w x tile_h, row stride in elements)
// from global memory into LDS.  Hand-packed D# per CDNA5 ISA:
//  group0: count=1 | lds_addr | global_addr[56:0] | type=2
//  group1: data_size=4B | tensor dims (huge: caller guarantees interior)
//          | tile_dim0/1 | tensor_dim0_stride
// 6-arg builtin form (clang-23 / therock toolchain): groups 2/3 + extra
// group are zero (2D tile, tile_dim2=0 disables higher dims).
// Caller must be the only wave issuing (uniform branch!) and must
// s_wait_tensorcnt(0) + barrier before consumers read LDS.
// ---------------------------------------------------------------------------
__device__ __forceinline__ void tdm_load_tile_f32(uint32_t lds_off, const float* g,
                                                  int tile_w, int tile_h, int stride) {
  uint64_t ga = (uint64_t)(uintptr_t)g;
  u32x4 g0;
  g0[0] = 1u;                                            // count=1, user mode
  g0[1] = lds_off;                                       // lds_addr (bytes)
  g0[2] = (unsigned)ga;                                  // global_addr[31:0]
  g0[3] = ((unsigned)(ga >> 32) & 0x01FFFFFFu)
          | 0x80000000u;                                 // addr[56:32] | type=2
  i32x8 g1;
  g1[0] = (int)(2u << 16);                               // data_size = 4 bytes
  g1[1] = 0;                                             // tensor_dim0=2^30 (lo16=0)
  g1[2] = 0x4000;                                        // dim0 hi | dim1=2^30 lo
  g1[3] = 0x4000 | (tile_w << 16);                       // dim1 hi | tile_dim0
  g1[4] = tile_h;                                        // tile_dim1 (tile_dim2=0)
  g1[5] = stride;                                        // tensor_dim0_stride lo32
  g1[6] = 0;
  g1[7] = 0;
  i32x4 gz4; gz4[0] = gz4[1] = gz4[2] = gz4[3] = 0;      // groups 2/3 unused (2D)
  i32x8 gz8;
  gz8[0] = gz8[1] = gz8[2] = gz8[3] = 0;
  gz8[4] = gz8[5] = gz8[6] = gz8[7] = 0;
  __builtin_amdgcn_tensor_load_to_lds(g0, g1, gz4, gz4, gz8, 0);
}

__device__ __forceinline__ bool wave0_uniform(int tid) {
  // readfirstlane -> SGPR -> guaranteed scalar branch (TDM ignores EXEC, so
  // we must *branch around* it in non-issuing waves, not mask it).
  return (__builtin_amdgcn_readfirstlane(tid) >> 5) == 0;
}

// ---------------------------------------------------------------------------
// Fused local normalized cross-correlation (zero padding), window = 2W x 2W.
// Per block: T x T outputs.  Separable box sums staged in LDS.
// FINAL=false: writes one_minus_rho = 1 - max(rho,-1).
// FINAL=true : reads one_minus_rho, computes mask & masked partial sum,
//              block-reduces, atomicAdd(double) to the accumulator.
// ---------------------------------------------------------------------------
template <int W, int T, bool FINAL>
__global__ __launch_bounds__(256) void xcorr_kernel(
    const float* __restrict__ Aimg, const float* __restrict__ Bimg,
    float* __restrict__ rho, double* __restrict__ acc) {
  constexpr int AREG = T + 4 * W - 2;   // raw input region
  constexpr int CREG = T + 2 * W - 1;   // centered-value region
  constexpr int CSTR = CREG + 1;
  constexpr int HSTR = T + 2;
  constexpr float AREA = 4.0f * (float)(W * W);

  __shared__ float rawA[AREG * AREG];   // TDM-packed (stride == AREG)
  __shared__ float rawB[AREG * AREG];
  __shared__ float hbuf[AREG * CSTR];   // horizontal box sums (reused A then B)
  __shared__ float cA[CREG * CSTR];
  __shared__ float cB[CREG * CSTR];
  __shared__ float hAA[CREG * HSTR];
  __shared__ float hBB[CREG * HSTR];
  __shared__ float hAB[CREG * HSTR];
  __shared__ float red[256];

  const int tid = threadIdx.x;
  const int r0 = blockIdx.y * T, c0 = blockIdx.x * T;
  const int gr = r0 - 2 * W, gc = c0 - 2 * W;

  const bool interior =
      (gr >= 0) && (gc >= 0) && (gr + AREG <= IMG_H) && (gc + AREG <= IMG_W);
  if (interior) {
    if (wave0_uniform(tid)) {
      tdm_load_tile_f32(lds_off_of(rawA), Aimg + (size_t)gr * IMG_W + gc,
                        AREG, AREG, IMG_W);
      tdm_load_tile_f32(lds_off_of(rawB), Bimg + (size_t)gr * IMG_W + gc,
                        AREG, AREG, IMG_W);
      __builtin_amdgcn_s_wait_tensorcnt(0);
    }
  } else {
    for (int idx = tid; idx < AREG * AREG; idx += 256) {
      int r = idx / AREG, c = idx % AREG;
      int rr = gr + r, cc = gc + c;
      bool in = (rr >= 0) && (cc >= 0) && (rr < IMG_H) && (cc < IMG_W);
      rawA[idx] = in ? Aimg[rr * IMG_W + cc] : 0.f;
      rawB[idx] = in ? Bimg[rr * IMG_W + cc] : 0.f;
    }
  }
  __syncthreads();

  // ---- A: horizontal box sums, then mu and centered values ----
  for (int idx = tid; idx < AREG * CREG; idx += 256) {
    int r = idx / CREG, c = idx % CREG;
    float s = 0.f;
#pragma unroll
    for (int q = 0; q < 2 * W; ++q) s += rawA[r * AREG + c + q];
    hbuf[r * CSTR + c] = s;
  }
  __syncthreads();
  for (int idx = tid; idx < CREG * CREG; idx += 256) {
    int i = idx / CREG, j = idx % CREG;
    float s = 0.f;
#pragma unroll
    for (int p = 0; p < 2 * W; ++p) s += hbuf[(i + p) * CSTR + j];
    int ar = r0 - W + i, ac = c0 - W + j;
    bool in = (ar >= 0) && (ac >= 0) && (ar < IMG_H) && (ac < IMG_W);
    cA[i * CSTR + j] = in ? (rawA[(i + W) * AREG + (j + W)] - s / AREA) : 0.f;
  }
  __syncthreads();

  // ---- B: same, reusing hbuf ----
  for (int idx = tid; idx < AREG * CREG; idx += 256) {
    int r = idx / CREG, c = idx % CREG;
    float s = 0.f;
#pragma unroll
    for (int q = 0; q < 2 * W; ++q) s += rawB[r * AREG + c + q];
    hbuf[r * CSTR + c] = s;
  }
  __syncthreads();
  for (int idx = tid; idx < CREG * CREG; idx += 256) {
    int i = idx / CREG, j = idx % CREG;
    float s = 0.f;
#pragma unroll
    for (int p = 0; p < 2 * W; ++p) s += hbuf[(i + p) * CSTR + j];
    int ar = r0 - W + i, ac = c0 - W + j;
    bool in = (ar >= 0) && (ac >= 0) && (ar < IMG_H) && (ac < IMG_W);
    cB[i * CSTR + j] = in ? (rawB[(i + W) * AREG + (j + W)] - s / AREA) : 0.f;
  }
  __syncthreads();

  // ---- horizontal box sums of the three products ----
  for (int idx = tid; idx < CREG * T; idx += 256) {
    int r = idx / T, j = idx % T;
    float saa = 0.f, sbb = 0.f, sab = 0.f;
#pragma unroll
    for (int q = 0; q < 2 * W; ++q) {
      float av = cA[r * CSTR + j + q], bv = cB[r * CSTR + j + q];
      saa += av * av;
      sbb += bv * bv;
      sab += av * bv;
    }
    hAA[r * HSTR + j] = saa;
    hBB[r * HSTR + j] = sbb;
    hAB[r * HSTR + j] = sab;
  }
  __syncthreads();

  // ---- vertical sums + normalized correlation ----
  float local = 0.f;
  for (int idx = tid; idx < T * T; idx += 256) {
    int i = idx / T, j = idx % T;
    float sii = 0.f, sjj = 0.f, sij = 0.f;
#pragma unroll
    for (int p = 0; p < 2 * W; ++p) {
      sii += hAA[(i + p) * HSTR + j];
      sjj += hBB[(i + p) * HSTR + j];
      sij += hAB[(i + p) * HSTR + j];
    }
    sii = fmaxf(sii, EPSF);
    sjj = fmaxf(sjj, EPSF);
    float r_ = sij / (sqrtf(sii * sjj) + EPSF);
    int o = (r0 + i) * IMG_W + (c0 + j);
    if (!FINAL) {
      rho[o] = 1.f - fmaxf(r_, -1.f);
    } else {
      float omr = rho[o];
      float omrm = 1.f - r_;
      local += (omr > omrm) ? omr : 0.f;
    }
  }

  if (FINAL) {
    red[tid] = local;
    __syncthreads();
    for (int s = 128; s > 0; s >>= 1) {
      if (tid < s) red[tid] += red[tid + s];
      __syncthreads();
    }
    if (tid == 0) atomicAdd(acc, (double)red[0]);
  }
}

// ---------------------------------------------------------------------------
// Separable 41-tap Gaussian blur via f32 WMMA banded GEMM.
// Horizontal pass: out(16x16 tile) = X(16x56) * G(56x16), G[p][j]=h[p-j].
// Block = 8 waves, each wave owns one 16x16 tile -> 16x128 outputs/block.
// Reflect padding applied while staging into LDS.
// ---------------------------------------------------------------------------
__global__ __launch_bounds__(256) void blur_h_kernel(
    const float* __restrict__ src, const float* __restrict__ k41,
    float* __restrict__ dst) {
  __shared__ float tile[16 * 169];  // rows r0..r0+15, cols c0-20..c0+147
  __shared__ float hoff[80];        // h[t] at index t+16, zero outside [0,41)

  const int tid = threadIdx.x;
  const int r0 = blockIdx.y * 16, c0 = blockIdx.x * 128;

  if (tid < 80) {
    int t = tid - 16;
    hoff[tid] = (t >= 0 && t < 41) ? sqrtf(fmaxf(k41[t * 41 + t], 0.f)) : 0.f;
  }
  for (int idx = tid; idx < 16 * 168; idx += 256) {
    int r = idx / 168, c = idx % 168;
    int cc = c0 - 20 + c;
    cc = (cc < 0) ? -cc : cc;                       // reflect (no edge repeat)
    cc = (cc > IMG_W - 1) ? (2 * IMG_W - 2 - cc) : cc;
    tile[r * 169 + c] = src[(r0 + r) * IMG_W + cc];
  }
  __syncthreads();

  const int lane = tid & 31, wave = tid >> 5;
  const int half = lane >> 4, n = lane & 15;
  const int wc = wave * 16;
  v8f acc = {0.f, 0.f, 0.f, 0.f, 0.f, 0.f, 0.f, 0.f};
#pragma unroll
  for (int s = 0; s < 14; ++s) {                    // K = 56 = 14 * 4
    int kb = s * 4 + half * 2;                      // A/B VGPR K layout: {0,2}/{1,3}
    v2f a, b;
    a.x = tile[n * 169 + wc + kb];                  // X[m=n][p=kb]
    a.y = tile[n * 169 + wc + kb + 1];
    b.x = hoff[kb - n + 16];                        // G[p=kb][j=n] = h[p-j]
    b.y = hoff[kb + 1 - n + 16];
    acc = wmma_f32_k4(a, b, acc);
  }
#pragma unroll
  for (int v = 0; v < 8; ++v)
    dst[(r0 + v + half * 8) * IMG_W + c0 + wc + n] = acc[v];
}

// Vertical pass: out(16x16) = Gv(16x56) * Y(56x16), Gv[i][p]=h[p-i].
// The 56x128 input tile is DMA'd to LDS by the TDM for interior blocks.
__global__ __launch_bounds__(256) void blur_v_kernel(
    const float* __restrict__ src, const float* __restrict__ k41,
    float* __restrict__ dst) {
  __shared__ float tile[56 * 128];  // rows r0-20..r0+35, cols c0..c0+127
  __shared__ float hoff[80];

  const int tid = threadIdx.x;
  const int r0 = blockIdx.y * 16, c0 = blockIdx.x * 128;

  if (tid < 80) {
    int t = tid - 16;
    hoff[tid] = (t >= 0 && t < 41) ? sqrtf(fmaxf(k41[t * 41 + t], 0.f)) : 0.f;
  }

  const bool interior = (r0 >= 20) && (r0 + 35 <= IMG_H - 1);
  if (interior) {
    if (wave0_uniform(tid)) {
      tdm_load_tile_f32(lds_off_of(tile), src + (size_t)(r0 - 20) * IMG_W + c0,
                        128, 56, IMG_W);
      __builtin_amdgcn_s_wait_tensorcnt(0);
    }
  } else {
    for (int idx = tid; idx < 56 * 128; idx += 256) {
      int r = idx >> 7, c = idx & 127;
      int rr = r0 - 20 + r;
      rr = (rr < 0) ? -rr : rr;
      rr = (rr > IMG_H - 1) ? (2 * IMG_H - 2 - rr) : rr;
      tile[idx] = src[rr * IMG_W + c0 + c];
    }
  }
  __syncthreads();

  const int lane = tid & 31, wave = tid >> 5;
  const int half = lane >> 4, n = lane & 15;
  const int wc = wave * 16;
  v8f acc = {0.f, 0.f, 0.f, 0.f, 0.f, 0.f, 0.f, 0.f};
#pragma unroll
  for (int s = 0; s < 14; ++s) {
    int kb = s * 4 + half * 2;
    v2f a, b;
    a.x = hoff[kb - n + 16];                        // Gv[m=n][p=kb] = h[p-m]
    a.y = hoff[kb + 1 - n + 16];
    b.x = tile[kb * 128 + wc + n];                  // Y[p=kb][j=n]
    b.y = tile[(kb + 1) * 128 + wc + n];
    acc = wmma_f32_k4(a, b, acc);
  }
#pragma unroll
  for (int v = 0; v < 8; ++v)
    dst[(r0 + v + half * 8) * IMG_W + c0 + wc + n] = acc[v];
}

__global__ void init_kernel(double* acc) { *acc = 0.0; }

__global__ void finalize_kernel(const double* acc, float* out) {
  out[0] = (float)(*acc * (1.0 / ((double)IMG_H * (double)IMG_W)));
}

// ---------------------------------------------------------------------------
extern "C" void kernel_launch(void* const* d_in, const int* in_sizes, int n_in,
                              void* d_out, int out_size, void* d_ws, size_t ws_size,
                              hipStream_t stream) {
  (void)in_sizes; (void)n_in; (void)out_size; (void)ws_size;
  const float* hrms = (const float*)d_in[0];
  const float* pan  = (const float*)d_in[1];
  const float* lrms = (const float*)d_in[2];
  const float* k41  = (const float*)d_in[3];
  float* out = (float*)d_out;

  const size_t IMG_BYTES = (size_t)IMG_H * IMG_W * sizeof(float);  // 16 MiB
  char* ws = (char*)d_ws;
  float*  rho    = (float*)(ws);
  float*  tmp1   = (float*)(ws + IMG_BYTES);
  float*  pan_lp = (float*)(ws + 2 * IMG_BYTES);
  double* acc    = (double*)(ws + 3 * IMG_BYTES);

  init_kernel<<<1, 1, 0, stream>>>(acc);

  dim3 gA(IMG_W / 32, IMG_H / 32);       // 64 x 64
  xcorr_kernel<2, 32, false><<<gA, 256, 0, stream>>>(hrms, pan, rho, acc);

  dim3 gB(IMG_W / 128, IMG_H / 16);      // 16 x 128
  blur_h_kernel<<<gB, 256, 0, stream>>>(pan, k41, tmp1);
  blur_v_kernel<<<gB, 256, 0, stream>>>(tmp1, k41, pan_lp);

  dim3 gD(IMG_W / 16, IMG_H / 16);       // 128 x 128
  xcorr_kernel<8, 16, true><<<gD, 256, 0, stream>>>(pan_lp, lrms, rho, acc);

  finalize_kernel<<<1, 1, 0, stream>>>(acc, out);
}